// TwoLayerGCN_4724464026100
// MI455X (gfx1250) — compile-verified
//
#include <hip/hip_runtime.h>
#include <hip/hip_bf16.h>

#define N_NODES 100000
#define DIM 64

typedef __attribute__((ext_vector_type(2))) float v2f;
typedef __attribute__((ext_vector_type(8))) float v8f;

// ---------------------------------------------------------------- utilities
__global__ void gcn_zero_f32(float* __restrict__ p, int n) {
    int i = blockIdx.x * blockDim.x + threadIdx.x;
    if (i < n) p[i] = 0.0f;
}

// one thread per edge: bump out-degree of src and in-degree of dst
__global__ void gcn_degrees(const int* __restrict__ esrc,
                            const int* __restrict__ edst,
                            float* __restrict__ deg_src,
                            float* __restrict__ deg_dst, int E) {
    int e = blockIdx.x * blockDim.x + threadIdx.x;
    if (e < E) {
        atomicAdd(&deg_src[esrc[e]], 1.0f);
        atomicAdd(&deg_dst[edst[e]], 1.0f);
    }
}

// deg^{-1/2}, 0-degree -> 0 (processes deg_src and deg_dst in one contiguous pass)
__global__ void gcn_norms(const float* __restrict__ deg,
                          float* __restrict__ norm, int n) {
    int i = blockIdx.x * blockDim.x + threadIdx.x;
    if (i < n) {
        float d = deg[i];
        norm[i] = (d > 0.0f) ? rsqrtf(d) : 0.0f;
    }
}

// 16 adjacent lanes cooperate on one edge: lane chunk c handles features [4c,4c+4)
// msg = x[src] * src_norm[src]; agg[dst] += msg  (fp32 atomics, L2-resident)
__global__ void gcn_scatter(const float* __restrict__ x,
                            const float* __restrict__ src_norm,
                            const int* __restrict__ esrc,
                            const int* __restrict__ edst,
                            float* __restrict__ agg, int E) {
    long long gid = (long long)blockIdx.x * blockDim.x + threadIdx.x;
    int e = (int)(gid >> 4);
    if (e >= E) return;
    int chunk = ((int)gid & 15) * 4;
    int s = esrc[e];
    int d = edst[e];
    float nv = src_norm[s];
    float4 v = *(const float4*)&x[(long long)s * DIM + chunk];
    float* ap = &agg[(long long)d * DIM + chunk];
    atomicAdd(ap + 0, v.x * nv);
    atomicAdd(ap + 1, v.y * nv);
    atomicAdd(ap + 2, v.z * nv);
    atomicAdd(ap + 3, v.w * nv);
}

// out[16b : 16b+16, :] = relu((dst_norm * agg) @ W + bias)
// 128 threads = 4 waves; wave w owns columns [16w, 16w+16).
// Uses V_WMMA_F32_16X16X4_F32, K swept 0..64 in steps of 4 (16 WMMAs/wave).
__global__ void __launch_bounds__(128)
gcn_gemm_bias_relu(const float* __restrict__ agg,
                   const float* __restrict__ dst_norm,
                   const float* __restrict__ W,
                   const float* __restrict__ bias,
                   float* __restrict__ out) {
    __shared__ float Ws[DIM * DIM];   // 16 KB
    __shared__ float As[16 * DIM];    // 4 KB, pre-scaled by dst_norm

    const int tid  = threadIdx.x;     // 0..127
    const int wave = tid >> 5;
    const int lane = tid & 31;
    const int row0 = blockIdx.x * 16;

    // cooperative stage of W (4096 floats, float4 granularity)
    for (int i = tid * 4; i < DIM * DIM; i += 128 * 4) {
        *(float4*)&Ws[i] = *(const float4*)&W[i];
    }
    // cooperative stage of the A tile, folding in dst_norm
    for (int i = tid * 4; i < 16 * DIM; i += 128 * 4) {
        int r = i >> 6;                // row within tile
        float nv = dst_norm[row0 + r];
        float4 v = *(const float4*)&agg[(long long)(row0 + r) * DIM + (i & 63)];
        v.x *= nv; v.y *= nv; v.z *= nv; v.w *= nv;
        *(float4*)&As[i] = v;
    }
    __syncthreads();

    const int col0 = wave * 16;
    const int half = lane >> 4;        // 0: K pair {0,1}; 1: K pair {2,3}
    const int l15  = lane & 15;        // M (for A) / N (for B) index

    v8f c = {};
#pragma unroll
    for (int k = 0; k < DIM; k += 4) {
        v2f a, b;
        // 32-bit A 16x4 layout: lanes 0-15 -> K0,K1 ; lanes 16-31 -> K2,K3
        a.x = As[l15 * DIM + k + 2 * half + 0];
        a.y = As[l15 * DIM + k + 2 * half + 1];
        // mirrored layout for B 4x16
        b.x = Ws[(k + 2 * half + 0) * DIM + col0 + l15];
        b.y = Ws[(k + 2 * half + 1) * DIM + col0 + l15];
        c = __builtin_amdgcn_wmma_f32_16x16x4_f32(
                /*neg_a=*/false, a, /*neg_b=*/false, b,
                /*c_mod=*/(short)0, c, /*reuse_a=*/false, /*reuse_b=*/false);
    }

    // C/D layout: VGPR j holds M = j (lanes 0-15) or M = j+8 (lanes 16-31), N = lane&15
#pragma unroll
    for (int j = 0; j < 8; ++j) {
        int row = row0 + j + half * 8;
        int col = col0 + l15;
        float v = c[j] + bias[col];
        out[(long long)row * DIM + col] = (v > 0.0f) ? v : 0.0f;
    }
}

// ---------------------------------------------------------------- launcher
extern "C" void kernel_launch(void* const* d_in, const int* in_sizes, int n_in,
                              void* d_out, int out_size, void* d_ws, size_t ws_size,
                              hipStream_t stream) {
    const float* x    = (const float*)d_in[0];
    const int*   esrc = (const int*)d_in[1];
    const int*   edst = (const int*)d_in[2];
    const float* W1   = (const float*)d_in[3];
    const float* b1   = (const float*)d_in[4];
    const float* W2   = (const float*)d_in[5];
    const float* b2   = (const float*)d_in[6];
    float* out = (float*)d_out;
    const int E = in_sizes[1];
    const int N = N_NODES;

    // workspace layout (floats):
    float* ws       = (float*)d_ws;
    float* deg_src  = ws;                 // N
    float* deg_dst  = ws + N;             // N
    float* norm_src = ws + 2LL * N;       // N
    float* norm_dst = ws + 3LL * N;       // N
    float* agg      = ws + 4LL * N;       // N*64
    float* h1       = ws + 4LL * N + (long long)N * DIM;  // N*64

    const int B = 256;

    // degrees + norms
    gcn_zero_f32<<<(2 * N + B - 1) / B, B, 0, stream>>>(deg_src, 2 * N);
    gcn_degrees<<<(E + B - 1) / B, B, 0, stream>>>(esrc, edst, deg_src, deg_dst, E);
    gcn_norms<<<(2 * N + B - 1) / B, B, 0, stream>>>(deg_src, norm_src, 2 * N);

    const long long nscatter = (long long)E * 16;
    const int scatter_blocks = (int)((nscatter + B - 1) / B);
    const int gemm_blocks = N / 16;  // 6250, exact

    // layer 1: agg = A_norm x ; h1 = relu(agg @ W1 + b1)
    gcn_zero_f32<<<(N * DIM + B - 1) / B, B, 0, stream>>>(agg, N * DIM);
    gcn_scatter<<<scatter_blocks, B, 0, stream>>>(x, norm_src, esrc, edst, agg, E);
    gcn_gemm_bias_relu<<<gemm_blocks, 128, 0, stream>>>(agg, norm_dst, W1, b1, h1);

    // layer 2: agg = A_norm h1 ; out = relu(agg @ W2 + b2)
    gcn_zero_f32<<<(N * DIM + B - 1) / B, B, 0, stream>>>(agg, N * DIM);
    gcn_scatter<<<scatter_blocks, B, 0, stream>>>(h1, norm_src, esrc, edst, agg, E);
    gcn_gemm_bias_relu<<<gemm_blocks, 128, 0, stream>>>(agg, norm_dst, W2, b2, out);
}